// Always_3822520893790
// MI455X (gfx1250) — compile-verified
//
#include <hip/hip_runtime.h>

// CDNA5 / gfx1250: wave32, WMMA 16x16xK. V_WMMA_F32_16X16X4_F32 computes the
// 16-wide causal window sum as a banded (0/1) matmul in full f32.
// exp/log use the raw base-2 hardware ops (v_exp_f32 / v_log_f32) via
// __builtin_amdgcn_* with pre-folded constants: one v_mul per transcendental.

typedef float v2f __attribute__((ext_vector_type(2)));
typedef float v8f __attribute__((ext_vector_type(8)));

#define B_DIM  16
#define T_DIM  8192
#define D_DIM  64
#define TILES_PER_WAVE 4
#define SUPER_T (T_DIM / (16 * TILES_PER_WAVE))   // 128 super-tiles along T
#define D_TILES (D_DIM / 16)                      // 4
// waves = 2(traces) * 16(B) * 128(superT) * 4(Dtiles) = 16384 -> 2048 blocks of 256

// exp(-5*x)      = exp2(EXP_C * x),   EXP_C = -5 * log2(e)
// -(1/5)*ln(s)   = LOG_C * log2(s),   LOG_C = -(1/5) * ln(2)
#define EXP_C (-7.213475204444817f)
#define LOG_C (-0.13862943611198906f)

__global__ __launch_bounds__(256)
void always_minish_wmma(const float* __restrict__ lower,
                        const float* __restrict__ upper,
                        float* __restrict__ out)
{
    const int lane = threadIdx.x & 31;
    const unsigned waveId = blockIdx.x * 8u + (threadIdx.x >> 5);

    const int dt    =  waveId        & (D_TILES - 1);
    const int stt   = (waveId >> 2)  & (SUPER_T - 1);
    const int b     = (waveId >> 9)  & (B_DIM   - 1);
    const int which =  waveId >> 13;

    const float* __restrict__ src = which ? upper : lower;
    float* __restrict__ dst = out + (size_t)which * ((size_t)B_DIM * T_DIM * D_DIM);

    const int half = lane >> 4;          // selects K={0,1} vs K={2,3} within a chunk
    const int col  = lane & 15;          // N column (d) for B/C, M row for A
    const int t0   = stt * (16 * TILES_PER_WAVE);
    const int d    = dt * 16 + col;

    const size_t base = (size_t)b * T_DIM * D_DIM + d;   // + row*D_DIM

    // ---- band weights (A matrix, 16x32 of 0/1), invariant across tiles ----
    // lane holds A[M, k] for M = col, k = c*4 + 2*half + {0,1}, c = 0..7
    const int M = col;
    v2f aw[8];
#pragma unroll
    for (int c = 0; c < 8; ++c) {
        int k0 = c * 4 + half * 2;
        aw[c].x = ((unsigned)(k0     - M - 1) < 16u) ? 1.0f : 0.0f;  // M < k <= M+16
        aw[c].y = ((unsigned)(k0 + 1 - M - 1) < 16u) ? 1.0f : 0.0f;
    }

    // ---- preload E = exp2(EXP_C*x) for rows tau = t0-16 .. t0+15 ----
    float e[16];
#pragma unroll
    for (int c = 0; c < 8; ++c) {
        int k0 = c * 4 + half * 2;
        int ra = t0 - 16 + k0;
        int rb = ra + 1;
        ra = ra < 0 ? 0 : ra;            // edge replication of row 0 (per-lane select)
        rb = rb < 0 ? 0 : rb;
        float xa = src[base + (size_t)ra * D_DIM];
        float xb = src[base + (size_t)rb * D_DIM];
        e[2 * c]     = __builtin_amdgcn_exp2f(EXP_C * xa);
        e[2 * c + 1] = __builtin_amdgcn_exp2f(EXP_C * xb);
    }

#pragma unroll
    for (int tile = 0; tile < TILES_PER_WAVE; ++tile) {
        // ---- banded matmul: C[16x16] = W(16x32 ones-band) x E(32x16) ----
        v8f acc = {};
#pragma unroll
        for (int c = 0; c < 8; ++c) {
            v2f bm;
            bm.x = e[2 * c];
            bm.y = e[2 * c + 1];
            acc = __builtin_amdgcn_wmma_f32_16x16x4_f32(
                false, aw[c], false, bm, (short)0, acc, false, false);
        }

        // ---- store: VGPR v -> row (t0 + tile*16 + v + 8*half), col d ----
        const size_t obase =
            ((size_t)b * T_DIM + (t0 + tile * 16 + half * 8)) * D_DIM + d;
#pragma unroll
        for (int v = 0; v < 8; ++v) {
            float r = LOG_C * __builtin_amdgcn_logf(acc[v]);
            __builtin_nontemporal_store(r, &dst[obase + (size_t)v * D_DIM]);
        }

        // ---- slide window by 16 rows: keep upper half of E, load 16 new rows ----
        if (tile < TILES_PER_WAVE - 1) {
#pragma unroll
            for (int i = 0; i < 8; ++i) e[i] = e[i + 8];
#pragma unroll
            for (int c = 4; c < 8; ++c) {
                // absolute row = (t0 + (tile+1)*16) - 16 + c*4 + 2*half  (>= 0 here)
                int row = t0 + tile * 16 + c * 4 + half * 2;
                float xa = src[base + (size_t)row * D_DIM];
                float xb = src[base + (size_t)(row + 1) * D_DIM];
                e[2 * c]     = __builtin_amdgcn_exp2f(EXP_C * xa);
                e[2 * c + 1] = __builtin_amdgcn_exp2f(EXP_C * xb);
            }
        }
    }
}

extern "C" void kernel_launch(void* const* d_in, const int* in_sizes, int n_in,
                              void* d_out, int out_size, void* d_ws, size_t ws_size,
                              hipStream_t stream) {
    (void)in_sizes; (void)n_in; (void)d_ws; (void)ws_size; (void)out_size;
    const float* lower = (const float*)d_in[0];
    const float* upper = (const float*)d_in[1];
    float* out = (float*)d_out;
    hipLaunchKernelGGL(always_minish_wmma, dim3(2048), dim3(256), 0, stream,
                       lower, upper, out);
}